// EDO_30434138259870
// MI455X (gfx1250) — compile-verified
//
#include <hip/hip_runtime.h>

// Lorenz-95 RK4, 200 steps, rows of 1024 fp32 sites, 16384 independent rows.
// One wave32 per row: each lane owns 32 contiguous sites in VGPRs; the 3 halo
// values per RK stage move via wave shuffles (ds_bpermute, no LDS, no barriers).
// Entire 200-step integration is register-resident: HBM touched exactly twice.

#define N_STEPS   200
#define DT        0.01f
#define FORCING   8.0f
#define ROWLEN    1024
#define NL        32            // sites per lane (ROWLEN / 32 lanes)
#define WPB       8             // waves per block
#define BLOCK     (WPB * 32)

// Tendency of Lorenz-95 on the 32-site chunk owned by this lane.
// v: stage state (registers). k: output tendency (registers).
__device__ __forceinline__ void l95_tendency(const float (&v)[NL], float (&k)[NL]) {
    const int lane  = threadIdx.x & 31;
    const int left  = (lane + 31) & 31;   // lane - 1, ring wrap
    const int right = (lane + 1)  & 31;   // lane + 1, ring wrap

    // Halo exchange: 3 cross-lane moves per stage (wave32 ds_bpermute path).
    const float lv30 = __shfl(v[NL - 2], left, 32);   // x_{g-2} for j=0
    const float lv31 = __shfl(v[NL - 1], left, 32);   // x_{g-1} for j=0, x_{g-2} for j=1
    const float rv0  = __shfl(v[0],      right, 32);  // x_{g+1} for j=31

    // dx_j = (x_{j+1} - x_{j-2}) * x_{j-1} - x_j + F  -> sub, sub, fma (3 VALU/site)
    k[0] = fmaf(v[1] - lv30, lv31, FORCING - v[0]);
    k[1] = fmaf(v[2] - lv31, v[0], FORCING - v[1]);
#pragma unroll
    for (int j = 2; j < NL - 1; ++j)
        k[j] = fmaf(v[j + 1] - v[j - 2], v[j - 1], FORCING - v[j]);
    k[NL - 1] = fmaf(rv0 - v[NL - 3], v[NL - 2], FORCING - v[NL - 1]);
}

__global__ void __launch_bounds__(BLOCK)
lorenz95_rk4_kernel(const float* __restrict__ in, float* __restrict__ out, int nrows) {
    const int wave = threadIdx.x >> 5;
    const int lane = threadIdx.x & 31;
    const int row  = blockIdx.x * WPB + wave;
    if (row >= nrows) return;   // whole wave exits together (row is wave-uniform)

    const float* __restrict__ src = in  + (size_t)row * ROWLEN + lane * NL;
    float*       __restrict__ dst = out + (size_t)row * ROWLEN + lane * NL;

    // gfx1250 prefetch path (global_prefetch_b8): pull the row toward the WGP.
    __builtin_prefetch(src, 0, 3);

    // Load 32 sites per lane as 8 x b128.
    float x[NL];
#pragma unroll
    for (int q = 0; q < NL / 4; ++q) {
        float4 t = reinterpret_cast<const float4*>(src)[q];
        x[4 * q + 0] = t.x; x[4 * q + 1] = t.y;
        x[4 * q + 2] = t.z; x[4 * q + 3] = t.w;
    }

    float k[NL], stg[NL], acc[NL];

#pragma unroll 1
    for (int s = 0; s < N_STEPS; ++s) {
        // k1: write tendency straight into acc (no register copies)
        l95_tendency(x, acc);
#pragma unroll
        for (int j = 0; j < NL; ++j)
            stg[j] = fmaf(0.5f * DT, acc[j], x[j]);
        // k2
        l95_tendency(stg, k);
#pragma unroll
        for (int j = 0; j < NL; ++j) {
            acc[j] = fmaf(2.0f, k[j], acc[j]);
            stg[j] = fmaf(0.5f * DT, k[j], x[j]);
        }
        // k3
        l95_tendency(stg, k);
#pragma unroll
        for (int j = 0; j < NL; ++j) {
            acc[j] = fmaf(2.0f, k[j], acc[j]);
            stg[j] = fmaf(DT, k[j], x[j]);
        }
        // k4 + state update: x += dt/6 * (acc + k4)
        l95_tendency(stg, k);
#pragma unroll
        for (int j = 0; j < NL; ++j)
            x[j] = fmaf(DT / 6.0f, acc[j] + k[j], x[j]);
    }

    // Store 8 x b128 per lane.
#pragma unroll
    for (int q = 0; q < NL / 4; ++q) {
        float4 t;
        t.x = x[4 * q + 0]; t.y = x[4 * q + 1];
        t.z = x[4 * q + 2]; t.w = x[4 * q + 3];
        reinterpret_cast<float4*>(dst)[q] = t;
    }
}

extern "C" void kernel_launch(void* const* d_in, const int* in_sizes, int n_in,
                              void* d_out, int out_size, void* d_ws, size_t ws_size,
                              hipStream_t stream) {
    const float* x = (const float*)d_in[0];
    float* out = (float*)d_out;
    const int nrows = in_sizes[0] / ROWLEN;            // 16384
    const int nblocks = (nrows + WPB - 1) / WPB;       // 2048 blocks of 8 waves
    lorenz95_rk4_kernel<<<nblocks, BLOCK, 0, stream>>>(x, out, nrows);
}